// Matcher_23381801959655
// MI455X (gfx1250) — compile-verified
//
#include <hip/hip_runtime.h>
#include <hip/hip_bf16.h>
#include <math.h>

#define NB 1024
#define PI_F 3.141592653f
#define TWO_PI_F 6.283185306f
#define IOU_THR_F 0.1f

typedef float v2f __attribute__((ext_vector_type(2)));
typedef float v8f __attribute__((ext_vector_type(8)));

// ---------------------------------------------------------------- CDNA5 async copy helpers

// LDS byte offset of a __shared__ object (generic -> LDS addrspace cast).
__device__ __forceinline__ unsigned lds_addr_of(const void* p) {
    return (unsigned)(uintptr_t)(const __attribute__((address_space(3))) char*)p;
}

// Per-lane 16B DMA global -> LDS, tracked by ASYNCcnt (no VGPR round trip).
__device__ __forceinline__ void async_ld_b128(unsigned lds_off, const void* g) {
    unsigned long long ga = (unsigned long long)(uintptr_t)g;
    asm volatile("global_load_async_to_lds_b128 %0, %1, off"
                 :: "v"(lds_off), "v"(ga) : "memory");
}

__device__ __forceinline__ void wait_async0() {
    asm volatile("s_wait_asynccnt 0x0" ::: "memory");
}

// ---------------------------------------------------------------- math utilities

__device__ __forceinline__ float limit_period_f(float v) {
    return v - floorf(v / TWO_PI_F + 0.5f) * TWO_PI_F;
}

__device__ __forceinline__ float cross2f(float ax, float ay, float bx, float by) {
    return ax * by - ay * bx;
}

__device__ __forceinline__ void rect_corners_f(const float* b, float* px, float* py) {
    float x = b[0], y = b[1], dx = b[3], dy = b[4], ang = b[6];
    float c = cosf(ang), s = sinf(ang);
    const float lx[4] = {0.5f, -0.5f, -0.5f, 0.5f};
    const float ly[4] = {0.5f, 0.5f, -0.5f, -0.5f};
#pragma unroll
    for (int l = 0; l < 4; ++l) {
        float vx = lx[l] * dx, vy = ly[l] * dy;
        px[l] = vx * c - vy * s + x;
        py[l] = vx * s + vy * c + y;
    }
}

// Sutherland-Hodgman clip of rect A against rect B (max 8 verts).
__device__ float bev_inter_area_f(const float* A, const float* B) {
    float pax[4], pay[4], pbx[4], pby[4];
    rect_corners_f(A, pax, pay);
    rect_corners_f(B, pbx, pby);
    float vx[8], vy[8];
#pragma unroll
    for (int l = 0; l < 8; ++l) { vx[l] = (l < 4) ? pax[l] : 0.f; vy[l] = (l < 4) ? pay[l] : 0.f; }
    int n = 4;
#pragma unroll
    for (int e = 0; e < 4; ++e) {
        float ax = pbx[e], ay = pby[e];
        float bx = pbx[(e + 1) & 3], by = pby[(e + 1) & 3];
        float dxe = bx - ax, dye = by - ay;
        float nvx[8], nvy[8];
#pragma unroll
        for (int l = 0; l < 8; ++l) { nvx[l] = 0.f; nvy[l] = 0.f; }
        int cnt = 0;
#pragma unroll
        for (int idx = 0; idx < 8; ++idx) {
            int prevI = (idx == 0) ? ((n - 1) & 7) : (idx - 1);
            float cvx = vx[idx], cvy = vy[idx];
            float pvx = vx[prevI], pvy = vy[prevI];
            float sp = cross2f(dxe, dye, cvx - ax, cvy - ay);
            float sq = cross2f(dxe, dye, pvx - ax, pvy - ay);
            bool inp = sp >= 0.f, inq = sq >= 0.f;
            float den = sq - sp;
            den = (fabsf(den) < 1e-9f) ? 1e-9f : den;
            float t = sq / den;
            float ix = pvx + t * (cvx - pvx);
            float iy = pvy + t * (cvy - pvy);
            bool live = idx < n;
            if (live && (inp != inq)) { if (cnt < 8) { nvx[cnt] = ix;  nvy[cnt] = iy;  } cnt++; }
            if (live && inp)          { if (cnt < 8) { nvx[cnt] = cvx; nvy[cnt] = cvy; } cnt++; }
        }
#pragma unroll
        for (int l = 0; l < 8; ++l) { vx[l] = nvx[l]; vy[l] = nvy[l]; }
        n = cnt;
    }
    if (n < 3) return 0.f;
    float area = 0.f;
#pragma unroll
    for (int idx = 0; idx < 8; ++idx) {
        if (idx < n) {
            int nxt = (idx == n - 1) ? 0 : ((idx + 1) & 7);
            area += cross2f(vx[idx], vy[idx], vx[nxt], vy[nxt]);
        }
    }
    return 0.5f * fabsf(area);
}

__device__ float iou3d_pair_f(const float* a, const float* b) {
    float ib  = bev_inter_area_f(a, b);
    float top = fminf(a[2] + a[5] * 0.5f, b[2] + b[5] * 0.5f);
    float bot = fmaxf(a[2] - a[5] * 0.5f, b[2] - b[5] * 0.5f);
    float h = fmaxf(top - bot, 0.f);
    float inter = ib * h;
    float va = a[3] * a[4] * a[5];
    float vb = b[3] * b[4] * b[5];
    return inter / fmaxf(va + vb - inter, 1e-6f);
}

__device__ __forceinline__ float block_reduce_sum256(float v, float* redf, int tid) {
    redf[tid] = v;
    __syncthreads();
    for (int st = 128; st > 0; st >>= 1) {
        if (tid < st) redf[tid] += redf[tid + st];
        __syncthreads();
    }
    float r = redf[0];
    __syncthreads();
    return r;
}

// ---------------------------------------------------------------- kernels

// Normalize yaw, build compact feature matrix Fc (NB x 8): [x,y,z,dx,dy,dz,sin,cos]
__global__ void prep_kernel(const float* __restrict__ boxes_in,
                            float* __restrict__ boxesn, float* __restrict__ dirs,
                            float* __restrict__ Fc) {
    int j = blockIdx.x * 256 + threadIdx.x;
    if (j >= NB) return;
    float b[7];
#pragma unroll
    for (int c = 0; c < 7; ++c) b[c] = boxes_in[j * 7 + c];
    b[6] = limit_period_f(b[6]);
#pragma unroll
    for (int c = 0; c < 7; ++c) boxesn[j * 7 + c] = b[c];
    dirs[j] = b[6];
#pragma unroll
    for (int c = 0; c < 6; ++c) Fc[j * 8 + c] = b[c];
    Fc[j * 8 + 6] = sinf(b[6]);
    Fc[j * 8 + 7] = cosf(b[6]);
}

// One block per row i; all 1024 boxes DMA'd into LDS via async-to-LDS (28 KB).
__global__ void iou_kernel(const float* __restrict__ boxesn, float* __restrict__ ious) {
    __shared__ __align__(16) float Bsh[NB * 7];
    int tid = threadIdx.x;
    __builtin_prefetch(boxesn + tid * 28, 0, 3);   // global_prefetch_b8 (L2 warm)
    for (int t = tid; t < (NB * 7) / 4; t += 256)
        async_ld_b128(lds_addr_of(&Bsh[t * 4]), boxesn + t * 4);
    wait_async0();
    __syncthreads();
    int i = blockIdx.x;
    float A[7];
#pragma unroll
    for (int c = 0; c < 7; ++c) A[c] = Bsh[i * 7 + c];
    for (int t = 0; t < 4; ++t) {
        int j = tid + t * 256;
        ious[(size_t)i * NB + j] = iou3d_pair_f(A, &Bsh[j * 7]);
    }
}

// Faithful serial greedy clustering (overwrite semantics preserved).
__global__ void cluster_kernel(const float* __restrict__ ious, int* __restrict__ ci) {
    __shared__ int ci_sh[NB];
    __shared__ int s_cur;
    int tid = threadIdx.x;
    ci_sh[tid] = 0;
    __syncthreads();
    int cid = 1;
    for (int it = 0; it < NB; ++it) {
        if (tid == 0) s_cur = 0x7fffffff;
        __syncthreads();
        if (ci_sh[tid] == 0) atomicMin(&s_cur, tid);
        __syncthreads();
        int cur = s_cur;
        bool has = (cur != 0x7fffffff);
        int curI = has ? cur : 0;
        float iou = ious[(size_t)curI * NB + tid];
        if (has && (iou > IOU_THR_F)) ci_sh[tid] = cid;
        if (has) cid++;
        __syncthreads();
    }
    ci[tid] = ci_sh[tid];
}

// Pack per-box A-operand data for the GEMM: (ci as float, score, dir, 0).
__global__ void pack_kernel(const int* __restrict__ ci, const float* __restrict__ scores,
                            const float* __restrict__ dirs, float4* __restrict__ P) {
    int j = blockIdx.x * 256 + threadIdx.x;
    if (j >= NB) return;
    float4 p;
    p.x = (float)ci[j];     // exact for values <= 2^24
    p.y = scores[j];
    p.z = dirs[j];
    p.w = 0.f;
    P[j] = p;
}

// Per-cluster scalars: score sum, ref direction (first argmax), flip vote, sorted-power score.
__global__ void fuse_pass1_kernel(const int* __restrict__ ci, const float* __restrict__ scores,
                                  const float* __restrict__ dirs,
                                  float* __restrict__ ssum_o, float* __restrict__ refd_o,
                                  float* __restrict__ flip_o, float* __restrict__ sf_o,
                                  float* __restrict__ valid_o) {
    __shared__ float s_sh[NB];
    __shared__ float redf[256];
    __shared__ int   redi[256];
    int tid = threadIdx.x;
    int label = blockIdx.x + 1;
#pragma unroll
    for (int t = 0; t < 4; ++t) {
        int j = tid + t * 256;
        s_sh[j] = (ci[j] == label) ? scores[j] : 0.f;
    }
    __syncthreads();

    float lsum = 0.f, bv = -1.f;
    int bi = 0;
#pragma unroll
    for (int t = 0; t < 4; ++t) {
        int j = tid + t * 256;
        float s = s_sh[j];
        lsum += s;
        if (s > bv) { bv = s; bi = j; }
    }
    float ssum = block_reduce_sum256(lsum, redf, tid);

    // first-occurrence argmax
    redf[tid] = bv; redi[tid] = bi;
    __syncthreads();
    for (int st = 128; st > 0; st >>= 1) {
        if (tid < st) {
            float ov = redf[tid + st]; int oi = redi[tid + st];
            if (ov > redf[tid] || (ov == redf[tid] && oi < redi[tid])) { redf[tid] = ov; redi[tid] = oi; }
        }
        __syncthreads();
    }
    int am = redi[0];
    __syncthreads();
    float ref = dirs[am];

    float lgt = 0.f;
#pragma unroll
    for (int t = 0; t < 4; ++t) {
        int j = tid + t * 256;
        float s = s_sh[j];
        if (s > 0.f) {
            float dd = fabsf(dirs[j] - ref);
            dd = (dd > PI_F) ? (TWO_PI_F - dd) : dd;
            if (dd > PI_F * 0.5f) lgt += s;
        }
    }
    float sc_gt = block_reduce_sum256(lgt, redf, tid);
    bool flip_gt = sc_gt <= (ssum - sc_gt);

    // sorted-power score via rank counting against LDS copy
    float lsf = 0.f;
    for (int t = 0; t < 4; ++t) {
        int j = tid + t * 256;
        float sj = s_sh[j];
        if (sj > 0.f) {
            int cnt = 0;
            for (int kk = 0; kk < NB; ++kk) {
                float sk = s_sh[kk];
                cnt += (sk > sj || (sk == sj && kk < j)) ? 1 : 0;
            }
            lsf += __powf(sj, (float)(cnt + 1));
        }
    }
    float sfsum = block_reduce_sum256(lsf, redf, tid);

    if (tid == 0) {
        ssum_o[blockIdx.x]  = ssum;
        refd_o[blockIdx.x]  = ref;
        flip_o[blockIdx.x]  = flip_gt ? 1.f : 0.f;
        sf_o[blockIdx.x]    = fminf(sfsum, 1.f);
        valid_o[blockIdx.x] = (ssum > 0.f) ? 1.f : 0.f;
    }
}

// One K-step (K advances by 4): two V_WMMA_F32_16X16X4_F32.
// All operands come from two unconditional ds_load_b128 + two ds_load_b32;
// masking is multiply-by-{0,1} so no load is predicated -> EXEC stays all-ones.
__device__ __forceinline__ void gemm_step(int k0, float labelf, float rden, float rr, bool flp,
                                          const float4* __restrict__ Psh,
                                          const float* __restrict__ Fsh,
                                          int mrow, v8f& d1, v8f& d2) {
    float4 p0 = Psh[k0];
    float4 p1 = Psh[k0 + 1];
    float m0 = (p0.x == labelf) ? 1.f : 0.f;
    float m1 = (p1.x == labelf) ? 1.f : 0.f;
    float w0 = p0.y * m0 * rden;
    float w1 = p1.y * m1 * rden;
    float dd0 = fabsf(p0.z - rr); dd0 = (dd0 > PI_F) ? (TWO_PI_F - dd0) : dd0;
    float dd1 = fabsf(p1.z - rr); dd1 = (dd1 > PI_F) ? (TWO_PI_F - dd1) : dd1;
    bool gt0 = dd0 > PI_F * 0.5f, gt1 = dd1 > PI_F * 0.5f;
    bool add0 = flp ? gt0 : (!gt0), add1 = flp ? gt1 : (!gt1);   // sin(d+pi*add)=±sin(d)
    v2f a1; a1.x = w0;               a1.y = w1;
    v2f a2; a2.x = add0 ? -w0 : w0;  a2.y = add1 ? -w1 : w1;
    int nn = mrow & 7;
    float f0 = Fsh[k0 * 8 + nn], f1 = Fsh[(k0 + 1) * 8 + nn];
    bool lo = mrow < 8;
    v2f b; b.x = lo ? f0 : 0.f; b.y = lo ? f1 : 0.f;
    d1 = __builtin_amdgcn_wmma_f32_16x16x4_f32(false, a1, false, b, (short)0, d1, false, false);
    d2 = __builtin_amdgcn_wmma_f32_16x16x4_f32(false, a2, false, b, (short)0, d2, false, false);
}

// Fused cluster GEMM: D1 = W @ F (centers), D2 = Wsigned @ F (circular-mean sin/cos).
// 8 waves/block, 16 clusters per wave, K unrolled x2 with split accumulators.
__global__ void fuse_gemm_kernel(const float* __restrict__ Fc, const float4* __restrict__ P,
                                 const float* __restrict__ ssum, const float* __restrict__ refd,
                                 const float* __restrict__ flip, float* __restrict__ acc) {
    __shared__ __align__(16) float  Fsh[NB * 8];  // 32 KB features
    __shared__ __align__(16) float4 Psh[NB];      // 16 KB packed (ci,score,dir)

    int tid = threadIdx.x;
    // Async DMA stage: Fc (2048 xfers), P (1024 xfers)
    for (int t = tid; t < (NB * 8) / 4; t += 256)
        async_ld_b128(lds_addr_of(&Fsh[t * 4]), Fc + t * 4);
    for (int t = tid; t < NB; t += 256)
        async_ld_b128(lds_addr_of(&Psh[t]), P + t);
    wait_async0();
    __builtin_amdgcn_s_wait_tensorcnt(0);
    __syncthreads();

    int lane  = tid & 31;                       // wave32
    int wave  = tid >> 5;                       // 0..7
    int mblk  = blockIdx.x * 128 + wave * 16;   // 16-cluster tile per wave
    int mrow  = lane & 15;                      // A: M = lane%16 ; B/C/D: N = lane%16
    int khalf = (lane >> 4) << 1;               // lanes 0-15 -> K 0,1 ; 16-31 -> K 2,3
    int m     = mblk + mrow;
    float labelf = (float)(m + 1);
    float sv  = ssum[m];
    float rden = 1.0f / ((sv > 0.f) ? sv : 1.f);
    float rr  = refd[m];
    bool flp  = flip[m] > 0.5f;

    v8f z = {0.f, 0.f, 0.f, 0.f, 0.f, 0.f, 0.f, 0.f};
    v8f d1a = z, d1b = z, d2a = z, d2b = z;

    for (int k = 0; k < NB; k += 8) {
        gemm_step(k + khalf,     labelf, rden, rr, flp, Psh, Fsh, mrow, d1a, d2a);
        gemm_step(k + 4 + khalf, labelf, rden, rr, flp, Psh, Fsh, mrow, d1b, d2b);
    }
    v8f d1 = d1a + d1b;
    v8f d2 = d2a + d2b;

    int half = (lane < 16) ? 0 : 8;             // C/D layout: VGPR r -> M = r (+8 hi lanes)
#pragma unroll
    for (int r = 0; r < 8; ++r) {
        int mm = mblk + r + half;
        acc[mm * 32 + mrow]      = d1[r];
        acc[mm * 32 + 16 + mrow] = d2[r];
    }
}

__global__ void finalize_kernel(const float* __restrict__ acc, const float* __restrict__ sf,
                                const float* __restrict__ validf, float* __restrict__ out) {
    int i = blockIdx.x * 256 + threadIdx.x;
    if (i >= NB) return;
    bool valid = validf[i] > 0.5f;
    float theta = atan2f(acc[i * 32 + 16 + 6], acc[i * 32 + 16 + 7]);
#pragma unroll
    for (int c = 0; c < 6; ++c) out[i * 7 + c] = valid ? acc[i * 32 + c] : 0.f;
    out[i * 7 + 6]  = valid ? theta : 0.f;
    out[NB * 7 + i] = valid ? sf[i] : 0.f;
    out[NB * 8 + i] = valid ? 1.f : 0.f;
}

// ---------------------------------------------------------------- launch

extern "C" void kernel_launch(void* const* d_in, const int* in_sizes, int n_in,
                              void* d_out, int out_size, void* d_ws, size_t ws_size,
                              hipStream_t stream) {
    (void)in_sizes; (void)n_in; (void)out_size; (void)ws_size;
    const float* det_boxes  = (const float*)d_in[0];
    const float* det_scores = (const float*)d_in[1];
    float* out = (float*)d_out;

    float* ws     = (float*)d_ws;
    float* ious   = ws;                            // NB*NB
    float* boxesn = ious + (size_t)NB * NB;        // NB*7   (16B aligned)
    float* dirs   = boxesn + NB * 7;               // NB     (16B aligned)
    float* Fc     = dirs + NB;                     // NB*8   (16B aligned)
    float* ssum   = Fc + NB * 8;                   // NB
    float* refd   = ssum + NB;                     // NB
    float* flip   = refd + NB;                     // NB
    float* sfv    = flip + NB;                     // NB
    float* validf = sfv + NB;                      // NB
    float* acc    = validf + NB;                   // NB*32
    float4* P     = (float4*)(acc + (size_t)NB * 32); // NB float4 (16B aligned)
    int*   ci     = (int*)(P + NB);                // NB ints

    prep_kernel      <<<NB / 256, 256, 0, stream>>>(det_boxes, boxesn, dirs, Fc);
    iou_kernel       <<<NB,       256, 0, stream>>>(boxesn, ious);
    cluster_kernel   <<<1,        NB,  0, stream>>>(ious, ci);
    pack_kernel      <<<NB / 256, 256, 0, stream>>>(ci, det_scores, dirs, P);
    fuse_pass1_kernel<<<NB,       256, 0, stream>>>(ci, det_scores, dirs, ssum, refd, flip, sfv, validf);
    fuse_gemm_kernel <<<8,        256, 0, stream>>>(Fc, P, ssum, refd, flip, acc);
    finalize_kernel  <<<NB / 256, 256, 0, stream>>>(acc, sfv, validf, out);
}